// GCN_55310588838369
// MI455X (gfx1250) — compile-verified
//
#include <hip/hip_runtime.h>
#include <hip/hip_bf16.h>

typedef float v2f __attribute__((ext_vector_type(2)));
typedef float v8f __attribute__((ext_vector_type(8)));

#define HC 16      // hidden channels
#define EPSV 1e-5f

// ---------------- utility ----------------
__global__ void zero_f32(float* __restrict__ p, int n) {
    int i = blockIdx.x * blockDim.x + threadIdx.x;
    if (i < n) p[i] = 0.0f;
}

// deg[dst] += 1 per edge (float atomics, L2-resident 400KB buffer)
__global__ void degree_kernel(const int* __restrict__ dst, float* __restrict__ deg, int E) {
    int e = blockIdx.x * blockDim.x + threadIdx.x;
    if (e < E) atomicAdd(&deg[dst[e]], 1.0f);
}

// dis = rsqrt(deg + 1)  (in place on deg buffer)
__global__ void rsqrt_kernel(float* __restrict__ d, int N) {
    int i = blockIdx.x * blockDim.x + threadIdx.x;
    if (i < N) d[i] = rsqrtf(d[i] + 1.0f);
}

// ---------------- WMMA GEMM: C[N x 16] = A[N x K] @ B[K x 16] ----------------
// one wave computes a 16x16 output tile using V_WMMA_F32_16X16X4_F32, K-loop step 4.
__global__ void gemm_wmma_f32(const float* __restrict__ A, const float* __restrict__ B,
                              float* __restrict__ C, int N, int K) {
    int wave = (int)((blockIdx.x * blockDim.x + threadIdx.x) >> 5);
    int lane = threadIdx.x & 31;
    int row0 = wave * 16;
    if (row0 >= N) return;                 // wave-uniform: EXEC stays all-ones for WMMA

    int m  = lane & 15;                    // row within tile (A) / col (B, C/D)
    int hi = lane >> 4;                    // half-wave selects K pair
    const float* arow = A + (size_t)(row0 + m) * K;

    v8f acc = {};
    for (int k0 = 0; k0 < K; k0 += 4) {
        int ka = k0 + 2 * hi;
        v2f a, b;
        a.x = arow[ka];
        a.y = arow[ka + 1];
        b.x = B[(size_t)ka * HC + m];      // B[k][n], row-major K x 16 (matches W layout)
        b.y = B[(size_t)(ka + 1) * HC + m];
        acc = __builtin_amdgcn_wmma_f32_16x16x4_f32(
            /*neg_a=*/false, a, /*neg_b=*/false, b,
            /*c_mod=*/(short)0, acc, /*reuse_a=*/false, /*reuse_b=*/false);
    }

    // D layout: VGPR r -> row (r + 8*hi), col m
    float* cbase = C + (size_t)(row0 + hi * 8) * HC + m;
#pragma unroll
    for (int r = 0; r < 8; ++r) cbase[(size_t)r * HC] = acc[r];
}

// ---------------- scatter phase ----------------
// out[i][c] = dis[i]^2 * hw[i][c] + b[c]   (self-loop contribution + bias)
__global__ void init_selfloop(const float* __restrict__ dis, const float* __restrict__ hw,
                              const float* __restrict__ bias, float* __restrict__ out, int NC) {
    int i = blockIdx.x * blockDim.x + threadIdx.x;
    if (i >= NC) return;
    int node = i >> 4, c = i & 15;
    float d = dis[node];
    out[i] = d * d * hw[i] + bias[c];
}

// out[dst] += dis[src]*dis[dst] * hw[src]  — 16 f32 atomics per edge, L2-resident target
__global__ void scatter_edges(const int* __restrict__ src, const int* __restrict__ dst,
                              const float* __restrict__ dis, const float* __restrict__ hw,
                              float* __restrict__ out, int E) {
    int e = blockIdx.x * blockDim.x + threadIdx.x;
    if (e >= E) return;
    int s = src[e], d = dst[e];
    float nrm = dis[s] * dis[d];
    const float4* hrow = (const float4*)(hw + (size_t)s * HC);   // 64B row, cacheline gather
    float* orow = out + (size_t)d * HC;
#pragma unroll
    for (int q = 0; q < 4; ++q) {
        float4 v = hrow[q];
        atomicAdd(orow + 4 * q + 0, nrm * v.x);
        atomicAdd(orow + 4 * q + 1, nrm * v.y);
        atomicAdd(orow + 4 * q + 2, nrm * v.z);
        atomicAdd(orow + 4 * q + 3, nrm * v.w);
    }
}

// ---------------- BatchNorm over relu(agg) ----------------
// stats[0..15] += sum relu(x) per channel; stats[16..31] += sum relu(x)^2
__global__ void bn_stats(const float* __restrict__ agg, float* __restrict__ stats, int N) {
    __shared__ float s1s[256];
    __shared__ float s2s[256];
    int c = threadIdx.x & 15;          // channel owned by this thread
    int g = threadIdx.x >> 4;          // row-group within block (16 groups)
    int stride = gridDim.x * 16;
    float s1 = 0.0f, s2 = 0.0f;
    for (int row = blockIdx.x * 16 + g; row < N; row += stride) {
        float v = agg[(size_t)row * HC + c];
        v = fmaxf(v, 0.0f);
        s1 += v;
        s2 += v * v;
    }
    s1s[threadIdx.x] = s1;
    s2s[threadIdx.x] = s2;
    __syncthreads();
    if (threadIdx.x < 16) {
        float t1 = 0.0f, t2 = 0.0f;
#pragma unroll
        for (int gg = 0; gg < 16; ++gg) {
            t1 += s1s[gg * 16 + threadIdx.x];
            t2 += s2s[gg * 16 + threadIdx.x];
        }
        atomicAdd(&stats[threadIdx.x], t1);
        atomicAdd(&stats[16 + threadIdx.x], t2);
    }
}

// stats -> scale (32..47), shift (48..63)
__global__ void bn_finalize(float* __restrict__ stats, const float* __restrict__ gamma,
                            const float* __restrict__ beta, float invN) {
    int c = threadIdx.x;
    if (c < 16) {
        float mu  = stats[c] * invN;
        float var = stats[16 + c] * invN - mu * mu;   // population var, matches jnp.var
        float sc  = rsqrtf(var + EPSV) * gamma[c];
        stats[32 + c] = sc;
        stats[48 + c] = beta[c] - mu * sc;
    }
}

// h = relu(agg) * scale + shift
__global__ void bn_apply(const float* __restrict__ agg, const float* __restrict__ stats,
                         float* __restrict__ h, int NC) {
    int i = blockIdx.x * blockDim.x + threadIdx.x;
    if (i >= NC) return;
    int c = i & 15;
    float v = fmaxf(agg[i], 0.0f);
    h[i] = v * stats[32 + c] + stats[48 + c];
}

// ---------------- head: logits = h @ Wfc + bfc, then log_softmax over 2 classes ----
__global__ void head_kernel(const float* __restrict__ h, const float* __restrict__ Wfc,
                            const float* __restrict__ bfc, float* __restrict__ out, int N) {
    int i = blockIdx.x * blockDim.x + threadIdx.x;
    if (i >= N) return;
    const float* r = h + (size_t)i * HC;
    float z0 = bfc[0], z1 = bfc[1];
#pragma unroll
    for (int c = 0; c < HC; ++c) {
        float v = r[c];
        z0 += v * Wfc[c * 2 + 0];
        z1 += v * Wfc[c * 2 + 1];
    }
    float mx  = fmaxf(z0, z1);
    float lse = mx + logf(__expf(z0 - mx) + __expf(z1 - mx));
    out[(size_t)i * 2 + 0] = z0 - lse;
    out[(size_t)i * 2 + 1] = z1 - lse;
}

// ---------------- driver ----------------
extern "C" void kernel_launch(void* const* d_in, const int* in_sizes, int n_in,
                              void* d_out, int out_size, void* d_ws, size_t ws_size,
                              hipStream_t stream) {
    const int IN = 128;
    const int N  = in_sizes[0] / IN;     // 100000
    const int E  = in_sizes[1] / 2;      // 3200000

    const float* x    = (const float*)d_in[0];
    const int*   ei   = (const int*)d_in[1];
    const int*   srcI = ei;              // edge_index[0]
    const int*   dstI = ei + E;          // edge_index[1]
    const float* W1   = (const float*)d_in[2];
    const float* b1   = (const float*)d_in[3];
    const float* W2   = (const float*)d_in[4];
    const float* b2   = (const float*)d_in[5];
    const float* W3   = (const float*)d_in[6];
    const float* b3   = (const float*)d_in[7];
    const float* g1   = (const float*)d_in[8];
    const float* be1  = (const float*)d_in[9];
    const float* g2   = (const float*)d_in[10];
    const float* be2  = (const float*)d_in[11];
    const float* Wfc  = (const float*)d_in[12];
    const float* bfc  = (const float*)d_in[13];

    float* ws   = (float*)d_ws;
    float* dis  = ws;                    // N
    float* bufA = dis + N;               // N*16 : hw (GEMM out)
    float* bufB = bufA + (size_t)N * HC; // N*16 : agg (scatter out)
    float* bufC = bufB + (size_t)N * HC; // N*16 : h (BN out)
    float* stats = bufC + (size_t)N * HC;// 64

    const int B256 = 256;
    const int NC = N * HC;
    dim3 blkN((N + B256 - 1) / B256);
    dim3 blkNC((NC + B256 - 1) / B256);
    dim3 blkE((E + B256 - 1) / B256);
    // 8 waves per 256-thread block -> 8 output tiles of 16 rows per block
    dim3 blkG((N / 16 + 7) / 8);
    float invN = 1.0f / (float)N;

    // degree / normalization
    zero_f32<<<blkN, B256, 0, stream>>>(dis, N);
    degree_kernel<<<blkE, B256, 0, stream>>>(dstI, dis, E);
    rsqrt_kernel<<<blkN, B256, 0, stream>>>(dis, N);

    // ---- layer 1: x(128) -> 16, relu + BN ----
    gemm_wmma_f32<<<blkG, B256, 0, stream>>>(x, W1, bufA, N, IN);
    init_selfloop<<<blkNC, B256, 0, stream>>>(dis, bufA, b1, bufB, NC);
    scatter_edges<<<blkE, B256, 0, stream>>>(srcI, dstI, dis, bufA, bufB, E);
    zero_f32<<<1, 64, 0, stream>>>(stats, 64);
    bn_stats<<<256, 256, 0, stream>>>(bufB, stats, N);
    bn_finalize<<<1, 16, 0, stream>>>(stats, g1, be1, invN);
    bn_apply<<<blkNC, B256, 0, stream>>>(bufB, stats, bufC, NC);

    // ---- layer 2: 16 -> 16, relu + BN ----
    gemm_wmma_f32<<<blkG, B256, 0, stream>>>(bufC, W2, bufA, N, HC);
    init_selfloop<<<blkNC, B256, 0, stream>>>(dis, bufA, b2, bufB, NC);
    scatter_edges<<<blkE, B256, 0, stream>>>(srcI, dstI, dis, bufA, bufB, E);
    zero_f32<<<1, 64, 0, stream>>>(stats, 64);
    bn_stats<<<256, 256, 0, stream>>>(bufB, stats, N);
    bn_finalize<<<1, 16, 0, stream>>>(stats, g2, be2, invN);
    bn_apply<<<blkNC, B256, 0, stream>>>(bufB, stats, bufC, NC);

    // ---- layer 3: 16 -> 16, no relu/BN ----
    gemm_wmma_f32<<<blkG, B256, 0, stream>>>(bufC, W3, bufA, N, HC);
    init_selfloop<<<blkNC, B256, 0, stream>>>(dis, bufA, b3, bufB, NC);
    scatter_edges<<<blkE, B256, 0, stream>>>(srcI, dstI, dis, bufA, bufB, E);

    // ---- head + log_softmax ----
    head_kernel<<<blkN, B256, 0, stream>>>(bufB, Wfc, bfc, (float*)d_out, N);
}